// sparse_normalization_59356448031141
// MI455X (gfx1250) — compile-verified
//
#include <hip/hip_runtime.h>
#include <stdint.h>

// ---------------------------------------------------------------------------
// Sparse symmetric-normalized Laplacian build:
//   pass 1: dedup edges via L2-resident open-addressing hash, count degrees
//           (edge stream staged through LDS with CDNA5 async global->LDS DMA,
//            double-buffered on ASYNCcnt)
//   pass 2: dis = rsqrt(deg) (0 for isolated)
//   pass 3: values/indices output, B128-vectorized, NT loads/stores so the
//           write-once 53 MB output stream does not evict the L2-resident
//           hash table (64 MB) + A (25.6 MB) working set (192 MB L2).
// Memory-bound: ~145 MB HBM traffic -> ~6.2us @ 23.3 TB/s. No dense matrix
// math in this op, so WMMA does not apply; CDNA5-specific paths exercised:
// global_load_async_to_lds_b32 + s_wait_asynccnt, global_prefetch_b8,
// TH=NT cache-policy loads/stores.
// ---------------------------------------------------------------------------

#define BLOCK_T 256

typedef int   v4i __attribute__((ext_vector_type(4)));   // native clang vectors:
typedef float v4f __attribute__((ext_vector_type(4)));   // legal for NT builtins

__device__ __forceinline__ unsigned long long mix64(unsigned long long x) {
  x += 0x9E3779B97F4A7C15ull;
  x = (x ^ (x >> 30)) * 0xBF58476D1CE4E5B9ull;
  x = (x ^ (x >> 27)) * 0x94D049BB133111EBull;
  return x ^ (x >> 31);
}

#if defined(__HIP_DEVICE_COMPILE__) && __has_builtin(__builtin_amdgcn_s_wait_asynccnt)
#define WAIT_ASYNC(n) __builtin_amdgcn_s_wait_asynccnt(n)
#else
#define WAIT_ASYNC(n) asm volatile("s_wait_asynccnt %0" ::"i"(n) : "memory")
#endif

__device__ __forceinline__ void async_load_b32(unsigned lds_byte_off, const int* gptr) {
  // GLOBAL_LOAD_ASYNC_TO_LDS_B32 (GV mode): VDST = per-lane LDS byte offset,
  // VADDR = 64-bit global address. Tracked by ASYNCcnt. Default TH (RT) so A
  // stays in L2 for the pass-3 re-read.
  asm volatile("global_load_async_to_lds_b32 %0, %1, off"
               :: "v"(lds_byte_off), "v"(gptr) : "memory");
}

// --------------------------- init ------------------------------------------
__global__ void init_ws_kernel(ulonglong2* __restrict__ hash2, unsigned H2,
                               int* __restrict__ deg, int N) {
  unsigned stride = gridDim.x * blockDim.x;
  unsigned i0 = blockIdx.x * blockDim.x + threadIdx.x;
  ulonglong2 empty; empty.x = ~0ull; empty.y = ~0ull;
  // Regular (RT) stores: the table is immediately re-read by pass-1 CAS probes.
  for (unsigned i = i0; i < H2; i += stride) hash2[i] = empty;   // B128 stores
  for (unsigned i = i0; i < (unsigned)N; i += stride) deg[i] = 0;
}

// --------------------------- pass 1: degrees -------------------------------
__device__ __forceinline__ void hash_insert_count(unsigned long long* __restrict__ hash,
                                                  unsigned Hmask, int* __restrict__ deg,
                                                  int r, int c, unsigned N) {
  unsigned long long key = (unsigned long long)(unsigned)r * N + (unsigned)c; // < N^2, != ~0
  unsigned h = (unsigned)mix64(key) & Hmask;
  for (;;) {
    unsigned long long prev = atomicCAS(&hash[h], ~0ull, key);
    if (prev == ~0ull) { atomicAdd(&deg[r], 1); break; } // first occurrence
    if (prev == key) break;                              // duplicate: skip
    h = (h + 1) & Hmask;                                 // linear probe
  }
}

__global__ void degree_kernel(const int* __restrict__ row, const int* __restrict__ col,
                              long long E, int N,
                              unsigned long long* __restrict__ hash, unsigned Hmask,
                              int* __restrict__ deg) {
  __shared__ int s_row[2][BLOCK_T];
  __shared__ int s_col[2][BLOCK_T];
  const int t = threadIdx.x;
  // Low 32 bits of the flat shared-space address are the LDS byte offset
  // (aperture check uses addr[63:32] only).
  unsigned lr[2], lc[2];
  lr[0] = (unsigned)(uintptr_t)(&s_row[0][t]);
  lr[1] = (unsigned)(uintptr_t)(&s_row[1][t]);
  lc[0] = (unsigned)(uintptr_t)(&s_col[0][t]);
  lc[1] = (unsigned)(uintptr_t)(&s_col[1][t]);

  const long long step = (long long)gridDim.x * BLOCK_T;
  long long base = (long long)blockIdx.x * BLOCK_T;

  // Double-buffered full tiles: block-uniform trip condition so the asynccnt
  // wait immediates are static. Prime buffer 0.
  int buf = 0;
  if (base + BLOCK_T <= E) {
    async_load_b32(lr[0], row + base + t);
    async_load_b32(lc[0], col + base + t);
  }
  while (base + BLOCK_T <= E) {
    long long nbase = base + step;
    bool have_next = (nbase + BLOCK_T <= E);   // uniform across the block
    if (have_next) {
      async_load_b32(lr[buf ^ 1], row + nbase + t);
      async_load_b32(lc[buf ^ 1], col + nbase + t);
      WAIT_ASYNC(2);     // current tile done; next tile's 2 ops still in flight
    } else {
      WAIT_ASYNC(0);
    }
    int r = s_row[buf][t];
    int c = s_col[buf][t];
    hash_insert_count(hash, Hmask, deg, r, c, (unsigned)N);
    buf ^= 1;
    base = nbase;
  }
  // Ragged tail (guarded, plain loads).
  for (long long e = base + t; e < E; e += step) {
    hash_insert_count(hash, Hmask, deg, row[e], col[e], (unsigned)N);
  }
}

// --------------------------- pass 2 ----------------------------------------
__global__ void deg_inv_sqrt_kernel(const int* __restrict__ deg,
                                    float* __restrict__ dis, int N) {
  int i = blockIdx.x * blockDim.x + threadIdx.x;
  if (i < N) {
    int d = deg[i];
    dis[i] = (d > 0) ? rsqrtf((float)d) : 0.0f;  // dis stays L2-resident (RT)
  }
}

// --------------------------- pass 3: outputs -------------------------------
__global__ void diag_kernel(float* __restrict__ vals, int* __restrict__ idx_row,
                            int* __restrict__ idx_col, int N) {
  int i = blockIdx.x * blockDim.x + threadIdx.x;
  if (i < N) {
    __builtin_nontemporal_store(1.0f, &vals[i]);   // write-once: TH=NT
    __builtin_nontemporal_store(i, &idx_row[i]);
    __builtin_nontemporal_store(i, &idx_col[i]);
  }
}

// 4 edges per lane: NT global_load_b128 for indices (last use of A), NT
// global_store_b128 for all three output streams so they bypass L2; dis[]
// gathers (0.4 MB, RT) hit L2.
__global__ void edge_out_vec_kernel(const v4i* __restrict__ row4,
                                    const v4i* __restrict__ col4,
                                    const float* __restrict__ dis,
                                    long long E4, long long N,
                                    float* __restrict__ vals,
                                    int* __restrict__ idx_row,
                                    int* __restrict__ idx_col) {
  const long long stride = (long long)gridDim.x * blockDim.x;
  for (long long i = (long long)blockIdx.x * blockDim.x + threadIdx.x; i < E4; i += stride) {
    __builtin_prefetch(row4 + i + stride, 0, 0);   // global_prefetch_b8
    __builtin_prefetch(col4 + i + stride, 0, 0);
    v4i r = __builtin_nontemporal_load(row4 + i);
    v4i c = __builtin_nontemporal_load(col4 + i);
    v4f v;
    v.x = -dis[r.x] * dis[c.x];
    v.y = -dis[r.y] * dis[c.y];
    v.z = -dis[r.z] * dis[c.z];
    v.w = -dis[r.w] * dis[c.w];
    long long o = N + 4 * i;            // 16B-aligned (N % 4 == 0 checked on host)
    __builtin_nontemporal_store(v, (v4f*)(vals + o));
    __builtin_nontemporal_store(r, (v4i*)(idx_row + o));
    __builtin_nontemporal_store(c, (v4i*)(idx_col + o));
  }
}

// Scalar fallback / tail: edges in [e0, E).
__global__ void edge_out_scalar_kernel(const int* __restrict__ row,
                                       const int* __restrict__ col,
                                       const float* __restrict__ dis,
                                       long long e0, long long E, long long N,
                                       float* __restrict__ vals,
                                       int* __restrict__ idx_row,
                                       int* __restrict__ idx_col) {
  const long long stride = (long long)gridDim.x * blockDim.x;
  for (long long e = e0 + (long long)blockIdx.x * blockDim.x + threadIdx.x; e < E; e += stride) {
    int r = __builtin_nontemporal_load(row + e);
    int c = __builtin_nontemporal_load(col + e);
    __builtin_nontemporal_store(-dis[r] * dis[c], &vals[N + e]);
    __builtin_nontemporal_store(r, &idx_row[N + e]);
    __builtin_nontemporal_store(c, &idx_col[N + e]);
  }
}

// --------------------------- host ------------------------------------------
extern "C" void kernel_launch(void* const* d_in, const int* in_sizes, int n_in,
                              void* d_out, int out_size, void* d_ws, size_t ws_size,
                              hipStream_t stream) {
  (void)n_in; (void)out_size;
  // d_in[0] = x (float32, N x 128) -- only its shape matters, never read.
  // d_in[1] = A (int32 after JAX x64-demotion), shape [2, E] flattened.
  const int D = 128;
  const long long N = (long long)in_sizes[0] / D;
  const long long E = (long long)in_sizes[1] / 2;
  const int* A = (const int*)d_in[1];
  const int* row = A;
  const int* col = A + E;

  // Output layout: lap_values float32 [N+E], then lap_indices int32 [2, N+E]
  // (concatenated flat in return order; indices stored as int32 bit patterns).
  float* vals = (float*)d_out;
  int* idx = (int*)(vals + (N + E));
  int* idx_row = idx;
  int* idx_col = idx + (N + E);

  // Workspace: deg[N] i32 | dis[N] f32 | hash[H] u64, H = pow2 (<= 8M entries
  // = 64 MB; with deg/dis and A it is L2-resident on MI455X's 192 MB L2).
  char* ws = (char*)d_ws;
  size_t degBytes = (((size_t)N * sizeof(int)) + 255) & ~(size_t)255;
  size_t disBytes = degBytes;
  int* deg = (int*)ws;
  float* dis = (float*)(ws + degBytes);
  unsigned long long* hash = (unsigned long long*)(ws + degBytes + disBytes);
  size_t remain = (ws_size > degBytes + disBytes) ? (ws_size - degBytes - disBytes) : 0;
  int Hlog = 16;
  while (Hlog < 23 && ((size_t)sizeof(unsigned long long) << (Hlog + 1)) <= remain) Hlog++;
  unsigned H = 1u << Hlog;

  const int B = BLOCK_T;
  const int gridN = (int)((N + B - 1) / B);
  const int gridE = 2048;   // grid-stride: 524288 lanes over 3.2M edges

  hipLaunchKernelGGL(init_ws_kernel, dim3(2048), dim3(B), 0, stream,
                     (ulonglong2*)hash, H / 2u, deg, (int)N);
  hipLaunchKernelGGL(degree_kernel, dim3(gridE), dim3(B), 0, stream,
                     row, col, E, (int)N, hash, H - 1u, deg);
  hipLaunchKernelGGL(deg_inv_sqrt_kernel, dim3(gridN), dim3(B), 0, stream, deg, dis, (int)N);
  hipLaunchKernelGGL(diag_kernel, dim3(gridN), dim3(B), 0, stream, vals, idx_row, idx_col, (int)N);

  // Vector path needs 16B alignment of the "+N" offsets and the base pointers.
  bool vec_ok = (N % 4 == 0) && (E % 4 == 0) &&
                (((uintptr_t)row & 15) == 0) && (((uintptr_t)col & 15) == 0) &&
                (((uintptr_t)vals & 15) == 0) && (((uintptr_t)idx & 15) == 0);
  if (vec_ok) {
    long long E4 = E / 4;
    hipLaunchKernelGGL(edge_out_vec_kernel, dim3(gridE), dim3(B), 0, stream,
                       (const v4i*)row, (const v4i*)col, dis, E4, N,
                       vals, idx_row, idx_col);
  } else {
    hipLaunchKernelGGL(edge_out_scalar_kernel, dim3(gridE), dim3(B), 0, stream,
                       row, col, dis, 0ll, E, N, vals, idx_row, idx_col);
  }
}